// CRPS_64639257805427
// MI455X (gfx1250) — compile-verified
//
#include <hip/hip_runtime.h>

typedef __attribute__((ext_vector_type(2))) float v2f;
typedef __attribute__((ext_vector_type(8))) float v8f;

#define WAVES_PER_BLOCK 8
#define BLOCKS 2048

#define LOG2E 1.4426950408889634f
#define LN2   0.69314718055994531f
#define INV_SQRT2 0.70710678118654752f

// Raw hardware ops: v_rcp_f32 / v_log_f32 (log2) / v_exp_f32 (exp2).
// All inputs in this kernel are safely normal, so no denorm guards needed.
__device__ __forceinline__ float fast_rcp(float x)  { return __builtin_amdgcn_rcpf(x); }
__device__ __forceinline__ float rawlog2(float x)   { return __builtin_amdgcn_logf(x); }
__device__ __forceinline__ float rawexp2(float x)   { return __builtin_amdgcn_exp2f(x); }

// q = 1 + erf(u), branch-free (Abramowitz-Stegun 7.1.26, |err| <= 1.5e-7).
__device__ __forceinline__ float one_plus_erf(float u) {
    float au = fabsf(u);
    float t  = fast_rcp(fmaf(0.3275911f, au, 1.0f));
    float p  = fmaf(t, 1.061405429f, -1.453152027f);
    p = fmaf(t, p, 1.421413741f);
    p = fmaf(t, p, -0.284496736f);
    p = fmaf(t, p, 0.254829592f);
    p *= t;
    float ex = rawexp2(-LOG2E * (au * au));
    float pe = p * ex;                      // = erfc(|u|)
    return (u >= 0.0f) ? (2.0f - pe) : pe;  // 1 + erf(u)
}

__global__ __launch_bounds__(256) void crps_main_kernel(
        const float* __restrict__ logits,   // [N,2] row-major: mu, pre_log_sigma
        const float* __restrict__ y_true,   // [N,2] row-major: t, e
        float* __restrict__ partials,       // [BLOCKS]
        int N) {
    const int lane      = threadIdx.x & 31;
    const int waveInBlk = threadIdx.x >> 5;
    const int wave      = blockIdx.x * WAVES_PER_BLOCK + waveInBlk;
    const int nWaves    = gridDim.x * WAVES_PER_BLOCK;
    const int half      = lane >> 4;     // 0: K slots {0,1}; 1: K slots {2,3}
    const int row       = lane & 15;     // matrix row owned by this lane
    const int T         = (N + 15) >> 4; // 16-element tiles

    const float H = 0.9999f / 31.0f;     // uniform step of linspace(1e-4, 1, 32)

    // Per-lane grid values + log2(grid) for the 16 k-indices this lane touches
    // (fixed for the whole kernel: k = 4*j + 2*half + s).
    float gk[16], lg2k[16];
#pragma unroll
    for (int j = 0; j < 8; ++j) {
#pragma unroll
        for (int s = 0; s < 2; ++s) {
            int k = 4 * j + 2 * half + s;
            float g = 1e-4f + (float)k * H;
            gk[2 * j + s]   = g;
            lg2k[2 * j + s] = rawlog2(g);
        }
    }

    const float2* __restrict__ lg2p = (const float2*)logits;
    const float2* __restrict__ yt2p = (const float2*)y_true;

    v2f bones; bones[0] = 1.0f; bones[1] = 1.0f;  // B = all ones (weights folded into A)
    v8f acc = {};                                  // WMMA accumulator, persists across tiles

    for (int tile = wave; tile < T; tile += nWaves) {
        int i = tile * 16 + row;
        float valid = (i < N) ? 1.0f : 0.0f;
        int ic = (i < N) ? i : (N - 1);

        float2 lv = lg2p[ic];
        float2 yv = yt2p[ic];
        float mu  = lv.x;
        float pls = lv.y;
        float t   = yv.x;
        float e   = yv.y;

        // sigma = exp(min(softplus(pls) - 0.5, 10)) + 1e-8   (log2-domain hardware ops)
        float u     = rawexp2(-LOG2E * fabsf(pls));
        float sp    = fmaxf(pls, 0.0f) + LN2 * rawlog2(1.0f + u);
        float sigma = rawexp2(LOG2E * fminf(sp - 0.5f, 10.0f)) + 1e-8f;

        float is  = fast_rcp(sigma) * INV_SQRT2;  // 1/(sigma*sqrt(2)), erf-arg scale
        float is2 = LN2 * is;                     // applied to log2 inputs

        float lt2 = rawlog2(t);                   // log2(z1) = lt2 + log2(g_k)
        float y2  = fast_rcp(t + 1e-4f);
        float la2 = rawlog2(t + 1e-4f);           // log2(z2) = log2(g_k) - la2

        // erf-arg = fma(log2(g_k), is2, B)  (one FMA per eval)
        float muis = mu * is;
        float B1 = fmaf(lt2, is2, -muis);         // ((ln z1) - mu)/(sigma*sqrt2)
        float B2 = fmaf(-la2, is2,  muis);        // ((ln z2) + mu)/(sigma*sqrt2)

        // Trapezoid scale dz = y*H, 0.25 from Phi^2 = 0.25*q^2, e gates integral 2.
        float c1 = valid * t  * (H * 0.25f);
        float c2 = valid * e * y2 * (H * 0.25f);

#pragma unroll
        for (int j = 0; j < 8; ++j) {
            v2f a1, a2;
#pragma unroll
            for (int s = 0; s < 2; ++s) {
                float g  = gk[2 * j + s];
                float lg = lg2k[2 * j + s];
                // trapezoid endpoint halving: k==0 (j0,s0,half0), k==31 (j7,s1,half1)
                float c1w = c1, c2w = c2;
                if (j == 0 && s == 0 && half == 0) { c1w *= 0.5f; c2w *= 0.5f; }
                if (j == 7 && s == 1 && half == 1) { c1w *= 0.5f; c2w *= 0.5f; }

                float q1 = one_plus_erf(fmaf(lg, is2, B1));
                a1[s] = q1 * q1 * c1w;

                float q2 = one_plus_erf(fmaf(lg, is2, B2));
                float r2 = fast_rcp(fmaf(y2, g, 1e-4f));   // g(z) = 1/(z+1e-4)
                a2[s] = q2 * q2 * (r2 * c2w);
            }
            // D[m,n] += sum_k A[m,k] * 1  -> row sums accumulate into acc
            acc = __builtin_amdgcn_wmma_f32_16x16x4_f32(
                      false, a1, false, bones, (short)0, acc, false, false);
            acc = __builtin_amdgcn_wmma_f32_16x16x4_f32(
                      false, a2, false, bones, (short)0, acc, false, false);
        }
    }

    // Each row's CRPS sum is replicated across 16 columns; divide by 16 at the end.
    float part = acc[0] + acc[1] + acc[2] + acc[3] + acc[4] + acc[5] + acc[6] + acc[7];
#pragma unroll
    for (int m = 16; m >= 1; m >>= 1)
        part += __shfl_xor(part, m, 32);

    __shared__ float sdata[WAVES_PER_BLOCK];
    if (lane == 0) sdata[waveInBlk] = part;
    __syncthreads();
    if (threadIdx.x == 0) {
        float s = 0.0f;
#pragma unroll
        for (int w = 0; w < WAVES_PER_BLOCK; ++w) s += sdata[w];
        partials[blockIdx.x] = s;
    }
}

__global__ __launch_bounds__(256) void crps_reduce_kernel(
        const float* __restrict__ partials, float* __restrict__ out,
        int nPart, float scale) {
    __shared__ float sdata[WAVES_PER_BLOCK];
    float s = 0.0f;
    for (int idx = threadIdx.x; idx < nPart; idx += blockDim.x) s += partials[idx];
#pragma unroll
    for (int m = 16; m >= 1; m >>= 1)
        s += __shfl_xor(s, m, 32);
    int lane = threadIdx.x & 31, wv = threadIdx.x >> 5;
    if (lane == 0) sdata[wv] = s;
    __syncthreads();
    if (threadIdx.x == 0) {
        float tot = 0.0f;
#pragma unroll
        for (int w = 0; w < WAVES_PER_BLOCK; ++w) tot += sdata[w];
        out[0] = tot * scale;
    }
}

extern "C" void kernel_launch(void* const* d_in, const int* in_sizes, int n_in,
                              void* d_out, int out_size, void* d_ws, size_t ws_size,
                              hipStream_t stream) {
    const float* logits = (const float*)d_in[0];
    const float* y_true = (const float*)d_in[1];
    int N = in_sizes[0] / 2;            // logits is [N,2]
    float* partials = (float*)d_ws;     // BLOCKS floats = 8 KB scratch

    crps_main_kernel<<<BLOCKS, 256, 0, stream>>>(logits, y_true, partials, N);

    // /16 for column replication in the WMMA D matrix, /N for the mean.
    float scale = 1.0f / (16.0f * (float)N);
    crps_reduce_kernel<<<1, 256, 0, stream>>>(partials, (float*)d_out, BLOCKS, scale);
}